// GCN_38517266710862
// MI455X (gfx1250) — compile-verified
//
#include <hip/hip_runtime.h>
#include <hip/hip_bf16.h>

typedef __attribute__((ext_vector_type(2))) float v2f;
typedef __attribute__((ext_vector_type(8))) float v8f;

// ---------------------------------------------------------------------------
// Degree / normalization precompute (shared across all 3 layers)
// ---------------------------------------------------------------------------
__global__ void k_deg_init(float* __restrict__ deg, int n) {
    int i = blockIdx.x * blockDim.x + threadIdx.x;
    if (i < n) deg[i] = 1.0f;  // self loop
}

__global__ void k_deg_count(const int* __restrict__ dst, float* __restrict__ deg, int e) {
    int i = blockIdx.x * blockDim.x + threadIdx.x;
    if (i < e) atomicAdd(&deg[dst[i]], 1.0f);
}

__global__ void k_norm(const float* __restrict__ deg, float* __restrict__ isq,
                       float* __restrict__ idg, int n) {
    int i = blockIdx.x * blockDim.x + threadIdx.x;
    if (i < n) {
        float d = deg[i];
        isq[i] = rsqrtf(d);
        idg[i] = 1.0f / d;
    }
}

__global__ void k_coef(const int* __restrict__ src, const int* __restrict__ dst,
                       const float* __restrict__ isq, float* __restrict__ coef, int e) {
    int i = blockIdx.x * blockDim.x + threadIdx.x;
    if (i < e) coef[i] = isq[src[i]] * isq[dst[i]];
}

// ---------------------------------------------------------------------------
// Pad x [N,5] -> x8 [N,8] (zero tail) so GEMM-1 loads are unconditional and
// 8-byte aligned.
// ---------------------------------------------------------------------------
__global__ void k_pad_x(const float* __restrict__ x, float* __restrict__ x8, int n) {
    long idx = (long)blockIdx.x * blockDim.x + threadIdx.x;
    if (idx < (long)n * 8) {
        int i = (int)(idx >> 3), j = (int)(idx & 7);
        x8[idx] = (j < 5) ? x[(long)i * 5 + j] : 0.0f;
    }
}

// ---------------------------------------------------------------------------
// Weight prep: Bt[col][k] = W[k][col] (transpose), zero-padded to KPAD x COLS.
// Makes the per-lane WMMA B operand a contiguous aligned b64 load.
// ---------------------------------------------------------------------------
__global__ void k_prep_w(const float* __restrict__ W, float* __restrict__ Bt,
                         int Kin, int Cin, int KPAD, int COLS) {
    int idx = blockIdx.x * blockDim.x + threadIdx.x;
    if (idx >= COLS * KPAD) return;
    int col = idx / KPAD, k = idx % KPAD;
    Bt[idx] = (k < Kin && col < Cin) ? W[k * Cin + col] : 0.0f;
}

// ---------------------------------------------------------------------------
// Dense GEMM via fp32 WMMA:  H[nrows, SC] = A[nrows, K] @ B[K, *]
//   A: row-major [nrows, K]   (K = 8 or 64, rows 8B-aligned)
//   Bt: transposed+padded weights [NT*16, K]
//   One wave per 16x16 output tile; NT waves per block (one per column tile).
//   K-loop fully unrolled: 2x global_load_b64 + v_wmma_f32_16x16x4_f32 per step.
//
// VGPR layouts (ISA 7.12.2, 32-bit data):
//   A 16x4 : lanes 0-15 -> {K=k,k+1} in a.x/a.y ; lanes 16-31 -> {K=k+2,k+3}
//   B 4x16 : mirrored                                         (b.x/b.y)
//   C/D    : VGPR r -> M=r (lanes 0-15) / M=8+r (lanes 16-31), N = lane&15
// nrows must be a multiple of 16 (100000 = 6250*16) so EXEC is all-1s.
// ---------------------------------------------------------------------------
template <int K, int NT, int SC>
__global__ void k_gemm_wmma(const float* __restrict__ A, const float* __restrict__ Bt,
                            float* __restrict__ H) {
    const int lane = threadIdx.x & 31;
    const int wave = threadIdx.x >> 5;   // column tile
    const int half = lane >> 4;          // 0: K pair {k,k+1}, 1: {k+2,k+3}
    const int l15  = lane & 15;
    const size_t row = (size_t)blockIdx.x * 16 + l15;
    const int    col = wave * 16 + l15;
    const float* Arow = A + row * K;
    const float* Bcol = Bt + (size_t)col * K;

    v8f c = {};
#pragma unroll
    for (int k = 0; k < K; k += 4) {
        const int ka = k + half * 2;
        v2f a = *(const v2f*)(Arow + ka);
        v2f b = *(const v2f*)(Bcol + ka);
        c = __builtin_amdgcn_wmma_f32_16x16x4_f32(
                /*neg_a=*/false, a, /*neg_b=*/false, b,
                /*c_mod=*/(short)0, c, /*reuse_a=*/false, /*reuse_b=*/false);
    }

    if (NT * 16 == SC || col < SC) {
        const size_t mbase = (size_t)blockIdx.x * 16 + half * 8;
#pragma unroll
        for (int r = 0; r < 8; ++r)
            H[(mbase + r) * SC + col] = c[r];
    }
}

// ---------------------------------------------------------------------------
// agg[i,j] = h[i,j] * inv_deg[i] + bias[j]   (self-loop term + bias)
// ---------------------------------------------------------------------------
__global__ void k_self_bias(const float* __restrict__ h, const float* __restrict__ idg,
                            const float* __restrict__ bias, float* __restrict__ agg,
                            long n_elems, int dim) {
    long idx = (long)blockIdx.x * blockDim.x + threadIdx.x;
    if (idx < n_elems) {
        int i = (int)(idx / dim);
        int j = (int)(idx % dim);
        agg[idx] = h[idx] * idg[i] + bias[j];
    }
}

// ---------------------------------------------------------------------------
// Edge scatter: agg[dst,:] += coef[e] * h[src,:]   (64-wide rows, f32 atomics)
// h (25.6 MB) is L2-resident, so gathers + no-return atomics stay in L2.
// ---------------------------------------------------------------------------
__global__ void k_scatter64(const int* __restrict__ src, const int* __restrict__ dst,
                            const float* __restrict__ coef, const float* __restrict__ h,
                            float* __restrict__ agg, int e) {
    int i = blockIdx.x * blockDim.x + threadIdx.x;
    if (i >= e) return;
    const int   s  = src[i];
    const int   d  = dst[i];
    const float cf = coef[i];
    const float4* hs = (const float4*)(h + (size_t)s * 64);
    float* ad = agg + (size_t)d * 64;
#pragma unroll
    for (int g = 0; g < 16; ++g) {
        float4 v = hs[g];
        atomicAdd(ad + 4 * g + 0, v.x * cf);
        atomicAdd(ad + 4 * g + 1, v.y * cf);
        atomicAdd(ad + 4 * g + 2, v.z * cf);
        atomicAdd(ad + 4 * g + 3, v.w * cf);
    }
}

__global__ void k_scatter2(const int* __restrict__ src, const int* __restrict__ dst,
                           const float* __restrict__ coef, const float* __restrict__ h,
                           float* __restrict__ out, int e) {
    int i = blockIdx.x * blockDim.x + threadIdx.x;
    if (i >= e) return;
    const int   s  = src[i];
    const int   d  = dst[i];
    const float cf = coef[i];
    float2 v = *(const float2*)(h + (size_t)s * 2);
    atomicAdd(out + (size_t)d * 2 + 0, v.x * cf);
    atomicAdd(out + (size_t)d * 2 + 1, v.y * cf);
}

__global__ void k_relu(float* __restrict__ x, long n) {
    long idx = (long)blockIdx.x * blockDim.x + threadIdx.x;
    if (idx < n) x[idx] = fmaxf(x[idx], 0.0f);
}

// ---------------------------------------------------------------------------
// Launch: deg/coef/weight prep once; 3 layers of (WMMA GEMM, self+bias init,
// edge scatter, relu).  Two ping-pong N*64 buffers (sequential stream order
// makes the reuse safe).  Layer 3 accumulates directly into d_out.
// ---------------------------------------------------------------------------
extern "C" void kernel_launch(void* const* d_in, const int* in_sizes, int n_in,
                              void* d_out, int out_size, void* d_ws, size_t ws_size,
                              hipStream_t stream) {
    const float* x  = (const float*)d_in[0];
    const int*   ei = (const int*)d_in[1];
    const float* W1 = (const float*)d_in[2];
    const float* b1 = (const float*)d_in[3];
    const float* W2 = (const float*)d_in[4];
    const float* b2 = (const float*)d_in[5];
    const float* W3 = (const float*)d_in[6];
    const float* b3 = (const float*)d_in[7];
    float* out = (float*)d_out;

    const int N = in_sizes[0] / 5;   // 100000 (multiple of 16)
    const int E = in_sizes[1] / 2;   // 3200000
    const int* src = ei;
    const int* dst = ei + E;

    float* ws   = (float*)d_ws;
    float* deg  = ws;                          // N
    float* isq  = deg  + N;                    // N
    float* idg  = isq  + N;                    // N
    float* coef = idg  + N;                    // E
    float* x8   = coef + E;                    // N*8   (padded input)
    float* w1t  = x8   + (size_t)N * 8;        // 64*8  (Bt for layer 1)
    float* w2t  = w1t  + 64 * 8;               // 64*64 (Bt for layer 2)
    float* w3t  = w2t  + 64 * 64;              // 16*64 (Bt for layer 3, padded)
    float* bufA = w3t  + 16 * 64;              // N*64
    float* bufB = bufA + (size_t)N * 64;       // N*64
    float* h3   = bufB + (size_t)N * 64;       // N*2

    const int T = 256;
    auto cdiv = [](long a, long b) { return (int)((a + b - 1) / b); };
    const long NH = (long)N * 64;

    // --- shared precompute ---
    k_deg_init <<<cdiv(N, T), T, 0, stream>>>(deg, N);
    k_deg_count<<<cdiv(E, T), T, 0, stream>>>(dst, deg, E);
    k_norm     <<<cdiv(N, T), T, 0, stream>>>(deg, isq, idg, N);
    k_coef     <<<cdiv(E, T), T, 0, stream>>>(src, dst, isq, coef, E);
    k_pad_x    <<<cdiv((long)N * 8, T), T, 0, stream>>>(x, x8, N);
    k_prep_w   <<<cdiv(64 * 8,  T), T, 0, stream>>>(W1, w1t, 5, 64, 8, 64);
    k_prep_w   <<<cdiv(64 * 64, T), T, 0, stream>>>(W2, w2t, 64, 64, 64, 64);
    k_prep_w   <<<cdiv(16 * 64, T), T, 0, stream>>>(W3, w3t, 64, 2, 64, 16);

    const int GT = cdiv(N, 16);  // 6250 row tiles

    // --- layer 1: h1 = x8 @ W1 ; agg = self+bias ; scatter ; relu ---
    k_gemm_wmma<8, 4, 64><<<GT, 128, 0, stream>>>(x8, w1t, bufA);
    k_self_bias<<<cdiv(NH, T), T, 0, stream>>>(bufA, idg, b1, bufB, NH, 64);
    k_scatter64<<<cdiv(E, T), T, 0, stream>>>(src, dst, coef, bufA, bufB, E);
    k_relu     <<<cdiv(NH, T), T, 0, stream>>>(bufB, NH);

    // --- layer 2 ---
    k_gemm_wmma<64, 4, 64><<<GT, 128, 0, stream>>>(bufB, w2t, bufA);
    k_self_bias<<<cdiv(NH, T), T, 0, stream>>>(bufA, idg, b2, bufB, NH, 64);
    k_scatter64<<<cdiv(E, T), T, 0, stream>>>(src, dst, coef, bufA, bufB, E);
    k_relu     <<<cdiv(NH, T), T, 0, stream>>>(bufB, NH);

    // --- layer 3: into d_out ---
    k_gemm_wmma<64, 1, 2><<<GT, 32, 0, stream>>>(bufB, w3t, h3);
    k_self_bias<<<cdiv((long)N * 2, T), T, 0, stream>>>(h3, idg, b3, out, (long)N * 2, 2);
    k_scatter2 <<<cdiv(E, T), T, 0, stream>>>(src, dst, coef, h3, out, E);
}